// GroundedSam2_20907900797176
// MI455X (gfx1250) — compile-verified
//
#include <hip/hip_runtime.h>
#include <hip/hip_bf16.h>
#include <stdint.h>

// ---------------------------------------------------------------------------
// Greedy NMS for N=8192 boxes on MI455X (gfx1250, wave32).
//   K1: single-WGP bitonic sort (scores desc, stable-ish tiebreak) in 64KB LDS
//   K2: suppression bitmask; union term area_i+area_j per 16x16 tile via
//       v_wmma_f32_16x16x4_f32 (rank-1 outer sum), intersection via VALU,
//       bits packed with wave32 __ballot; row boxes staged with async->LDS.
//   K3: chunked sequential greedy scan (512 x 16-bit chunks, 256 threads)
//   K4: scatter outputs (bbox*k | cat/-1 | conf*k | keep) in return order
// ---------------------------------------------------------------------------

#define NBOX    8192
#define IOU_THR 0.5f
#define CHUNKS  512      // 16-bit column chunks per row
#define WORDS32 256      // 32-bit words covering 8192 bits

typedef float v2f __attribute__((ext_vector_type(2)));
typedef float v8f __attribute__((ext_vector_type(8)));
typedef int   v4i __attribute__((ext_vector_type(4)));

// Builtin wants: (v4i __device__* gaddr, v4i __shared__* ldsaddr, imm, imm)
typedef __attribute__((address_space(1))) v4i as1_v4i;
typedef __attribute__((address_space(3))) v4i as3_v4i;

__device__ __forceinline__ void async_copy_b128(const void* g, void* l) {
#if __has_builtin(__builtin_amdgcn_global_load_async_to_lds_b128)
  __builtin_amdgcn_global_load_async_to_lds_b128((as1_v4i*)g, (as3_v4i*)l, 0, 0);
#else
  *(float4*)l = *(const float4*)g;   // fallback: plain load + ds store
#endif
}

__device__ __forceinline__ void wait_async0() {
#if __has_builtin(__builtin_amdgcn_s_wait_asynccnt)
  __builtin_amdgcn_s_wait_asynccnt(0);
#else
  asm volatile("s_wait_asynccnt 0" ::: "memory");
#endif
}

// ---------------------------------------------------------------------------
// K1: bitonic sort of 8192 (score,idx) pairs in LDS, one 1024-thread block.
// Produces order[p] (original index at sorted pos p), gathered sorted boxes,
// and per-box areas.
// ---------------------------------------------------------------------------
__global__ __launch_bounds__(1024) void nms_sort_kernel(
    const float* __restrict__ boxes, const float* __restrict__ scores,
    float* __restrict__ sortedBoxes, float* __restrict__ sortedArea,
    int* __restrict__ order) {
  __shared__ float sKey[NBOX];   // 32 KB
  __shared__ int   sIdx[NBOX];   // 32 KB
  const int t = threadIdx.x;

  for (int i = t; i < NBOX; i += 1024) { sKey[i] = scores[i]; sIdx[i] = i; }
  __syncthreads();

  for (int k = 2; k <= NBOX; k <<= 1) {
    for (int j = k >> 1; j > 0; j >>= 1) {
      for (int a = t; a < NBOX; a += 1024) {
        const int b = a ^ j;
        if (b > a) {
          const float ka = sKey[a], kb = sKey[b];
          const int   ia = sIdx[a], ib = sIdx[b];
          // "a comes first" = higher score, tie -> lower original index
          const bool aFirst = (ka > kb) || (ka == kb && ia < ib);
          const bool up = ((a & k) == 0);
          if (up ? !aFirst : aFirst) {
            sKey[a] = kb; sKey[b] = ka;
            sIdx[a] = ib; sIdx[b] = ia;
          }
        }
      }
      __syncthreads();
    }
  }

  for (int p = t; p < NBOX; p += 1024) {
    const int o = sIdx[p];
    order[p] = o;
    const float x1 = boxes[o * 4 + 0], y1 = boxes[o * 4 + 1];
    const float x2 = boxes[o * 4 + 2], y2 = boxes[o * 4 + 3];
    sortedBoxes[p * 4 + 0] = x1;
    sortedBoxes[p * 4 + 1] = y1;
    sortedBoxes[p * 4 + 2] = x2;
    sortedBoxes[p * 4 + 3] = y2;
    sortedArea[p] = (x2 - x1) * (y2 - y1);
  }
}

// ---------------------------------------------------------------------------
// K2: suppression mask. Block = 256 threads = 8 waves; each wave owns a
// 16x16 tile (16 sorted rows x 16 sorted cols). One WMMA computes the
// pairwise area-sum S[m][n] = area_row[m] + area_col[n]; VALU does the
// min/max intersection; ballot packs suppress bits; lanes 0..15 store one
// uint16 chunk per row: bit c set <=> row suppresses col (col > row).
// ---------------------------------------------------------------------------
__global__ __launch_bounds__(256) void nms_mask_kernel(
    const float* __restrict__ sortedBoxes, const float* __restrict__ sortedArea,
    unsigned short* __restrict__ mask16) {
  __shared__ float4   sRow[8][16];   // per-wave staged row boxes
  __shared__ unsigned sBal[8 * 8];   // per-wave ballot words

  const int tid   = threadIdx.x;
  const int ww    = tid >> 5;        // wave in block (0..7)
  const int lane  = tid & 31;
  const int hl    = lane >> 4;       // half of wave (0/1)
  const int ln    = lane & 15;       // M (for A) and N (for B/C/D)
  const int chunk = blockIdx.x;      // 0..511 -> 16 columns
  const int j0    = chunk * 16;
  const int i0    = blockIdx.y * 128 + ww * 16;
  const float4* boxes4 = (const float4*)sortedBoxes;

  // CDNA5 async global->LDS stage of this wave's 16 row boxes (16B each)
  if (lane < 16) async_copy_b128(&boxes4[i0 + lane], &sRow[ww][lane]);
  wait_async0();
  __syncthreads();

  // A (16x4 f32): row m = [area_row[m], 1, 0, 0]
  // B (4x16 f32): col n = [1, area_col[n], 0, 0]^T
  // lanes 0-15 hold K=0,1 in the two VGPRs; lanes 16-31 hold K=2,3 (zeros)
  v2f afrag, bfrag;
  if (hl == 0) {
    afrag.x = sortedArea[i0 + ln]; afrag.y = 1.0f;
    bfrag.x = 1.0f;                bfrag.y = sortedArea[j0 + ln];
  } else {
    afrag.x = 0.0f; afrag.y = 0.0f;
    bfrag.x = 0.0f; bfrag.y = 0.0f;
  }
  v8f acc = {};
  // D[m][n] = area_row[m] + area_col[n]  (exact union term before -inter)
  acc = __builtin_amdgcn_wmma_f32_16x16x4_f32(false, afrag, false, bfrag,
                                              (short)0, acc, false, false);

  const float4 cb  = boxes4[j0 + ln];   // this lane's fixed column box
  const int    col = j0 + ln;

#pragma unroll
  for (int v = 0; v < 8; ++v) {
    // C/D layout: VGPR v, lanes 0-15 -> M=v ; lanes 16-31 -> M=v+8
    const float4 rb = sRow[ww][v + 8 * hl];
    const float xx1 = fmaxf(rb.x, cb.x), yy1 = fmaxf(rb.y, cb.y);
    const float xx2 = fminf(rb.z, cb.z), yy2 = fminf(rb.w, cb.w);
    const float inter = fmaxf(xx2 - xx1, 0.0f) * fmaxf(yy2 - yy1, 0.0f);
    const float uni   = acc[v] - inter;         // area_r + area_c - inter
    const int   row   = i0 + v + 8 * hl;
    const bool  sup   = (uni > 0.0f) && (inter > IOU_THR * uni) && (row < col);
    const unsigned bal = (unsigned)__ballot(sup);  // wave32: 32 valid bits
    if (lane == 0) sBal[ww * 8 + v] = bal;
  }
  asm volatile("s_wait_dscnt 0" ::: "memory");  // lane0 LDS stores -> wave reads

  if (lane < 16) {
    // ballot v: bits[15:0] = row i0+v, bits[31:16] = row i0+v+8 (N = bit idx)
    const unsigned b = sBal[ww * 8 + (lane & 7)];
    const unsigned short val = (lane < 8) ? (unsigned short)(b & 0xffffu)
                                          : (unsigned short)(b >> 16);
    mask16[(size_t)(i0 + lane) * CHUNKS + chunk] = val;
  }
}

// ---------------------------------------------------------------------------
// K3: chunked greedy scan. One 256-thread block. 512 outer steps:
//   lane0 resolves 16 candidates bit-serially inside the chunk (only the
//   chunk's own mask word matters intra-chunk), then all 256 threads OR the
//   kept rows' full mask into the removed bitset in parallel.
// ---------------------------------------------------------------------------
__global__ __launch_bounds__(WORDS32) void nms_scan_kernel(
    const unsigned short* __restrict__ mask16, unsigned* __restrict__ keepWords) {
  __shared__ unsigned removed[WORDS32];
  __shared__ unsigned keepw[WORDS32];
  __shared__ unsigned curKeep;
  const int t = threadIdx.x;
  removed[t] = 0u;
  keepw[t]   = 0u;
  __syncthreads();

  for (int c = 0; c < CHUNKS; ++c) {
    const int w = c >> 1, h = c & 1;
    if (t == 0) {
      unsigned rem = (removed[w] >> (16 * h)) & 0xffffu;
      unsigned kb = 0u;
      for (int b = 0; b < 16; ++b) {
        if (!((rem >> b) & 1u)) {
          kb |= 1u << b;
          const int i = c * 16 + b;
          rem |= (unsigned)mask16[(size_t)i * CHUNKS + c];  // intra-chunk bits
        }
      }
      curKeep = kb;
      keepw[w]   |= kb  << (16 * h);
      removed[w] |= rem << (16 * h);
    }
    __syncthreads();

    unsigned kb = curKeep;
    unsigned r  = removed[t];
    while (kb) {
      const int b = __ffs(kb) - 1;
      kb &= kb - 1u;
      const int i = c * 16 + b;
      r |= (unsigned)mask16[(size_t)i * CHUNKS + 2 * t]
         | ((unsigned)mask16[(size_t)i * CHUNKS + 2 * t + 1] << 16);
    }
    removed[t] = r;
    __syncthreads();
  }
  keepWords[t] = keepw[t];
}

// ---------------------------------------------------------------------------
// K4: scatter back to original order and emit the 7N-float output:
//   [0,4N) bbox*k | [4N,5N) cat or -1 | [5N,6N) conf*k | [6N,7N) keep
// ---------------------------------------------------------------------------
__global__ __launch_bounds__(256) void nms_finalize_kernel(
    const float* __restrict__ boxes, const float* __restrict__ scores,
    const int* __restrict__ cats, const int* __restrict__ order,
    const unsigned* __restrict__ keepWords, float* __restrict__ out) {
  const int p = blockIdx.x * blockDim.x + threadIdx.x;
  if (p >= NBOX) return;
  const int orig = order[p];
  const unsigned kb = (keepWords[p >> 5] >> (p & 31)) & 1u;
  const float k = kb ? 1.0f : 0.0f;
  out[orig * 4 + 0] = boxes[orig * 4 + 0] * k;
  out[orig * 4 + 1] = boxes[orig * 4 + 1] * k;
  out[orig * 4 + 2] = boxes[orig * 4 + 2] * k;
  out[orig * 4 + 3] = boxes[orig * 4 + 3] * k;
  out[4 * NBOX + orig] = kb ? (float)cats[orig] : -1.0f;
  out[5 * NBOX + orig] = scores[orig] * k;
  out[6 * NBOX + orig] = k;
}

// ---------------------------------------------------------------------------
extern "C" void kernel_launch(void* const* d_in, const int* in_sizes, int n_in,
                              void* d_out, int out_size, void* d_ws, size_t ws_size,
                              hipStream_t stream) {
  (void)in_sizes; (void)n_in; (void)out_size; (void)ws_size;
  const float* boxes  = (const float*)d_in[0];  // [8192,4]
  const float* scores = (const float*)d_in[1];  // [8192]
  const int*   cats   = (const int*)d_in[2];    // [8192]
  float* out = (float*)d_out;                   // 7*8192 floats

  // Workspace layout (needs ~8.6 MB; all regions fully rewritten each call):
  char* ws = (char*)d_ws;
  float*          sortedBoxes = (float*)(ws);                        // 131072 B
  float*          sortedArea  = (float*)(ws + 131072);               //  32768 B
  int*            order       = (int*)  (ws + 163840);               //  32768 B
  unsigned short* mask16      = (unsigned short*)(ws + 196608);      // 8 MB
  unsigned*       keepWords   = (unsigned*)(ws + 196608 + 8388608);  // 1 KB

  nms_sort_kernel<<<1, 1024, 0, stream>>>(boxes, scores, sortedBoxes,
                                          sortedArea, order);
  nms_mask_kernel<<<dim3(CHUNKS, NBOX / 128), 256, 0, stream>>>(
      sortedBoxes, sortedArea, mask16);
  nms_scan_kernel<<<1, WORDS32, 0, stream>>>(mask16, keepWords);
  nms_finalize_kernel<<<NBOX / 256, 256, 0, stream>>>(boxes, scores, cats,
                                                      order, keepWords, out);
}